// TriPlane_6021544149405
// MI455X (gfx1250) — compile-verified
//
#include <hip/hip_runtime.h>
#include <hip/hip_bf16.h>

// TriPlane 6-grid bilinear gather. Memory-bound: ~1.9 GB traffic vs 0.58 GFLOP.
// WMMA does not apply (per-point batched 1x4 * 4x16 with point-dependent B
// operand is not expressible as a wave-level matmul). Optimization targets:
// coalesced 64B gathers (16 lanes = 16 features of one point), NT cache hints
// on streamed input/output so the 194 MB of grids stay resident in the 192 MB L2.

constexpr int kHX = 64;
constexpr int kHY = 64;
constexpr int kU  = 512;
constexpr int kV  = 512;
constexpr int kL  = 16;

__device__ __forceinline__ float blerp(float c11, float c21, float c12, float c22,
                                       float ar, float br) {
  // top = c11*(1-ar) + c21*ar ; bot = c12*(1-ar) + c22*ar ; res = top*(1-br)+bot*br
  float top = fmaf(ar, c21 - c11, c11);
  float bot = fmaf(ar, c22 - c12, c12);
  return fmaf(br, bot - top, top);
}

// Gather the 4 corner values (at feature lane f) of one plane.
#define GATHER4(G, I, J, a1, a2, b1, b2, c11, c21, c12, c22)        \
  {                                                                 \
    int mb = m * (I) * (J);                                         \
    c11 = (G)[(mb + (a1) * (J) + (b1)) * kL + f];                   \
    c21 = (G)[(mb + (a2) * (J) + (b1)) * kL + f];                   \
    c12 = (G)[(mb + (a1) * (J) + (b2)) * kL + f];                   \
    c22 = (G)[(mb + (a2) * (J) + (b2)) * kL + f];                   \
  }

__global__ __launch_bounds__(256) void triplane_gather_kernel(
    const int*   __restrict__ mptr,
    const float* __restrict__ hptr,
    const float* __restrict__ uptr,
    const float* __restrict__ vptr,
    const float* __restrict__ Fxy,
    const float* __restrict__ Fxu,
    const float* __restrict__ Fxv,
    const float* __restrict__ Fyu,
    const float* __restrict__ Fyv,
    const float* __restrict__ Fuv,
    float*       __restrict__ out,
    int N)
{
  int t = blockIdx.x * 256 + (int)threadIdx.x;
  int n = t >> 4;      // point index (2 points per wave32)
  int f = t & 15;      // feature lane 0..15 -> contiguous 64B per half-wave
  if (n >= N) return;

  // Streamed per-point inputs: non-temporal (do not pollute L2, grids live there).
  int   m  = __builtin_nontemporal_load(mptr + n);
  float hx = __builtin_nontemporal_load(hptr + 2 * n + 0);
  float hy = __builtin_nontemporal_load(hptr + 2 * n + 1);
  float uu = __builtin_nontemporal_load(uptr + n);
  float vv = __builtin_nontemporal_load(vptr + n);

  float ix = ((hx + 1.0f) * 0.5f) * (float)kHX; if (ix == (float)kHX) ix = (float)(kHX - 1);
  float iy = ((hy + 1.0f) * 0.5f) * (float)kHY; if (iy == (float)kHY) iy = (float)(kHY - 1);
  float iu = uu * (float)kU;                    if (iu == (float)kU)  iu = (float)(kU - 1);
  float iv = vv * (float)kV;                    if (iv == (float)kV)  iv = (float)(kV - 1);

  // All values non-negative: trunc == floor. Dims are powers of two: % -> &.
  int x1 = (int)ix; int x2 = (x1 + 1) & (kHX - 1); float xr = ix - (float)x1;
  int y1 = (int)iy; int y2 = (y1 + 1) & (kHY - 1); float yr = iy - (float)y1;
  int u1 = (int)iu; int u2 = (u1 + 1) & (kU  - 1); float ur = iu - (float)u1;
  int v1 = (int)iv; int v2 = (v1 + 1) & (kV  - 1); float vr = iv - (float)v1;

  // Issue all 24 gathers before any arithmetic (24 outstanding vmem loads per
  // thread -> deep MLP). Fuv (128 MB grid, worst L2 locality) first.
  float F0, F1, F2, F3, A0, A1, A2, A3, B0, B1, B2, B3;
  float C0, C1, C2, C3, D0, D1, D2, D3, E0, E1, E2, E3;
  GATHER4(Fuv, kU,  kV,  u1, u2, v1, v2, F0, F1, F2, F3);
  GATHER4(Fxy, kHX, kHY, x1, x2, y1, y2, A0, A1, A2, A3);
  GATHER4(Fxu, kHX, kU,  x1, x2, u1, u2, B0, B1, B2, B3);
  GATHER4(Fxv, kHX, kV,  x1, x2, v1, v2, C0, C1, C2, C3);
  GATHER4(Fyu, kHY, kU,  y1, y2, u1, u2, D0, D1, D2, D3);
  GATHER4(Fyv, kHY, kV,  y1, y2, v1, v2, E0, E1, E2, E3);

  // Output row n is 96 contiguous floats; half-wave writes 64B segments.
  // Non-temporal stores: 384 MB streamed once, never reread -> keep L2 for grids.
  float* o = out + (size_t)n * 96 + f;
  __builtin_nontemporal_store(blerp(A0, A1, A2, A3, xr, yr), o + 0);
  __builtin_nontemporal_store(blerp(B0, B1, B2, B3, xr, ur), o + 16);
  __builtin_nontemporal_store(blerp(C0, C1, C2, C3, xr, vr), o + 32);
  __builtin_nontemporal_store(blerp(D0, D1, D2, D3, yr, ur), o + 48);
  __builtin_nontemporal_store(blerp(E0, E1, E2, E3, yr, vr), o + 64);
  __builtin_nontemporal_store(blerp(F0, F1, F2, F3, ur, vr), o + 80);
}

extern "C" void kernel_launch(void* const* d_in, const int* in_sizes, int n_in,
                              void* d_out, int out_size, void* d_ws, size_t ws_size,
                              hipStream_t stream) {
  // Input order: r, m, h, u, v, Fxy, Fxu, Fxv, Fyu, Fyv, Fuv.  (r is unused.)
  const int    N    = in_sizes[0];
  const int*   m    = (const int*)  d_in[1];
  const float* h    = (const float*)d_in[2];
  const float* u    = (const float*)d_in[3];
  const float* v    = (const float*)d_in[4];
  const float* Fxy  = (const float*)d_in[5];
  const float* Fxu  = (const float*)d_in[6];
  const float* Fxv  = (const float*)d_in[7];
  const float* Fyu  = (const float*)d_in[8];
  const float* Fyv  = (const float*)d_in[9];
  const float* Fuv  = (const float*)d_in[10];
  float*       out  = (float*)d_out;

  const int threads = 256;               // 8 wave32 per workgroup
  const long long total = (long long)N * 16;
  const int blocks = (int)((total + threads - 1) / threads);
  triplane_gather_kernel<<<blocks, threads, 0, stream>>>(
      m, h, u, v, Fxy, Fxu, Fxv, Fyu, Fyv, Fuv, out, N);
}